// GNNMolTailEncoder_9251359555633
// MI455X (gfx1250) — compile-verified
//
#include <hip/hip_runtime.h>
#include <hip/hip_bf16.h>

// Problem constants (match reference)
#define NN 50000
#define EE 300000
#define DD 256
#define D2 512
#define LL 5
#define VV 8

// mlp tiling
#define TM 32        // node rows per workgroup (2 row-tiles of 16)
#define SZP 260      // LDS pitch for z tile   (260 % 64 == 4 -> conflict-free col reads)
#define ZMP 516      // LDS pitch for mid tile (516 % 64 == 4)

typedef float v2f __attribute__((ext_vector_type(2)));
typedef float v8f __attribute__((ext_vector_type(8)));

#if defined(__has_builtin)
#if __has_builtin(__builtin_amdgcn_wmma_f32_16x16x4_f32)
#define USE_WMMA_F32X4 1
#endif
#endif

__device__ __forceinline__ v8f wmma_f32x4(v2f a, v2f b, v8f c) {
#ifdef USE_WMMA_F32X4
  return __builtin_amdgcn_wmma_f32_16x16x4_f32(false, a, false, b, (short)0, c,
                                               false, false);
#else
  c[0] += a.x * b.x + a.y * b.y;  // compile-safe placeholder
  return c;
#endif
}

// -------- z = (1 + eps[l]) * h  (vectorized) ------------------------------
__global__ __launch_bounds__(256) void init_z_kernel(const float4* __restrict__ h,
                                                     const float* __restrict__ eps,
                                                     int l, float4* __restrict__ z) {
  size_t i = (size_t)blockIdx.x * 256 + threadIdx.x;
  if (i < (size_t)NN * DD / 4) {
    float s = 1.0f + eps[l];
    float4 v = h[i];
    v.x *= s; v.y *= s; v.z *= s; v.w *= s;
    z[i] = v;
  }
}

// -------- edge phase: z[dst] += relu(h[src] + bond_emb lookups) -----------
__global__ __launch_bounds__(256) void edge_kernel(const float* __restrict__ h,
                                                   const int* __restrict__ src,
                                                   const int* __restrict__ dst,
                                                   const int* __restrict__ ea,
                                                   const float* __restrict__ be0,
                                                   const float* __restrict__ be1,
                                                   const float* __restrict__ be2,
                                                   float* __restrict__ z) {
  int e = blockIdx.x * 8 + (threadIdx.x >> 5);   // one wave32 per edge
  if (e >= EE) return;
  int lane = threadIdx.x & 31;
  int s = src[e];
  int t = dst[e];
  int a0 = ea[3 * e + 0];
  int a1 = ea[3 * e + 1];
  int a2 = ea[3 * e + 2];
  const float4* hs = (const float4*)(h + (size_t)s * DD);
  const float4* p0 = (const float4*)(be0 + (size_t)a0 * DD);
  const float4* p1 = (const float4*)(be1 + (size_t)a1 * DD);
  const float4* p2 = (const float4*)(be2 + (size_t)a2 * DD);
  float* zt = z + (size_t)t * DD;
#pragma unroll
  for (int j = 0; j < 2; ++j) {
    int q = lane + 32 * j;                       // float4 index (64 per row)
    float4 m = hs[q], e0 = p0[q], e1 = p1[q], e2 = p2[q];
    float vx = fmaxf(m.x + e0.x + e1.x + e2.x, 0.0f);
    float vy = fmaxf(m.y + e0.y + e1.y + e2.y, 0.0f);
    float vz = fmaxf(m.z + e0.z + e1.z + e2.z, 0.0f);
    float vw = fmaxf(m.w + e0.w + e1.w + e2.w, 0.0f);
    atomicAdd(&zt[4 * q + 0], vx);               // global_atomic_add_f32
    atomicAdd(&zt[4 * q + 1], vy);
    atomicAdd(&zt[4 * q + 2], vz);
    atomicAdd(&zt[4 * q + 3], vw);
  }
}

// -------- fused MLP: (z@W1+b1)->BN->relu->(@W2+b2)->BN->(relu)->+h --------
__global__ __launch_bounds__(256) void mlp_kernel(const float* __restrict__ z,
                                                  float* __restrict__ h,   // in-place residual update
                                                  const float* __restrict__ W1,
                                                  const float* __restrict__ b1,
                                                  const float* __restrict__ g1,
                                                  const float* __restrict__ s1,
                                                  const float* __restrict__ W2,
                                                  const float* __restrict__ b2,
                                                  const float* __restrict__ g2,
                                                  const float* __restrict__ s2,
                                                  int relu_out) {
  __shared__ float sZ[TM][SZP];      // 32 x 256 activation tile  (~33 KB)
  __shared__ float sMid[TM][ZMP];    // 32 x 512 intermediate     (~66 KB)
  const float inv = 0.99999500003749973f;  // 1/sqrt(1 + 1e-5)

  const int tid = threadIdx.x;
  const int row0 = blockIdx.x * TM;

  // ---- async-copy z tile into LDS (b128 granules, ASYNCcnt tracked) ------
  // 32 rows x 64 float4-chunks = 2048 chunks, 8 per thread.
  for (int j = 0; j < 8; ++j) {
    int i = tid + 256 * j;
    int r = i >> 6;
    int c4 = (i & 63) * 4;
    int grow = row0 + r;
    if (grow < NN) {
      unsigned lds_off = (unsigned)(uintptr_t)(&sZ[r][c4]);
      const float* gp = z + (size_t)grow * DD + c4;
      asm volatile("global_load_async_to_lds_b128 %0, %1, off"
                   :: "v"(lds_off), "v"(gp) : "memory");
    } else {
      sZ[r][c4 + 0] = 0.0f; sZ[r][c4 + 1] = 0.0f;
      sZ[r][c4 + 2] = 0.0f; sZ[r][c4 + 3] = 0.0f;
    }
  }
  asm volatile("s_wait_asynccnt 0x0" ::: "memory");
  __syncthreads();

  const int wave = tid >> 5;
  const int lane = tid & 31;
  const int half = lane >> 4;     // 0: K+0/1, acc rows 0-7; 1: K+2/3, acc rows 8-15
  const int lrow = lane & 15;
  const int koff = 2 * half;

  // ---- GEMM1: [32,256] @ [256,512] -> sMid -------------------------------
  // wave owns cols [wave*64, wave*64+63] (4 col-tiles) x 2 row-tiles.
  {
    v8f acc[2][4];
#pragma unroll
    for (int rt = 0; rt < 2; ++rt)
#pragma unroll
      for (int t = 0; t < 4; ++t)
        acc[rt][t] = (v8f){0.f, 0.f, 0.f, 0.f, 0.f, 0.f, 0.f, 0.f};
#pragma unroll 4
    for (int k0 = 0; k0 < DD; k0 += 4) {
      v2f a0, a1;
      a0.x = sZ[lrow][k0 + koff];        a0.y = sZ[lrow][k0 + koff + 1];
      a1.x = sZ[lrow + 16][k0 + koff];   a1.y = sZ[lrow + 16][k0 + koff + 1];
      const float* wrow = W1 + (size_t)(k0 + koff) * D2 + wave * 64 + lrow;
#pragma unroll
      for (int t = 0; t < 4; ++t) {
        v2f b;
        b.x = wrow[t * 16];
        b.y = wrow[t * 16 + D2];
        acc[0][t] = wmma_f32x4(a0, b, acc[0][t]);
        acc[1][t] = wmma_f32x4(a1, b, acc[1][t]);
      }
    }
#pragma unroll
    for (int t = 0; t < 4; ++t) {
      const int col = wave * 64 + t * 16 + lrow;
      const float bia = b1[col];
      const float gg = g1[col] * inv;
      const float ss = s1[col];
#pragma unroll
      for (int rt = 0; rt < 2; ++rt)
#pragma unroll
        for (int r = 0; r < 8; ++r) {
          float v = (acc[rt][t][r] + bia) * gg + ss;
          sMid[rt * 16 + r + 8 * half][col] = fmaxf(v, 0.0f);
        }
    }
  }
  __syncthreads();

  // ---- GEMM2: [32,512] @ [512,256] -> h (+BN, relu?, residual) -----------
  // wave owns cols [wave*32, wave*32+31] (2 col-tiles) x 2 row-tiles.
  {
    v8f acc[2][2];
#pragma unroll
    for (int rt = 0; rt < 2; ++rt)
#pragma unroll
      for (int t = 0; t < 2; ++t)
        acc[rt][t] = (v8f){0.f, 0.f, 0.f, 0.f, 0.f, 0.f, 0.f, 0.f};
#pragma unroll 4
    for (int k0 = 0; k0 < D2; k0 += 4) {
      v2f a0, a1;
      a0.x = sMid[lrow][k0 + koff];       a0.y = sMid[lrow][k0 + koff + 1];
      a1.x = sMid[lrow + 16][k0 + koff];  a1.y = sMid[lrow + 16][k0 + koff + 1];
      const float* wrow = W2 + (size_t)(k0 + koff) * DD + wave * 32 + lrow;
#pragma unroll
      for (int t = 0; t < 2; ++t) {
        v2f b;
        b.x = wrow[t * 16];
        b.y = wrow[t * 16 + DD];
        acc[0][t] = wmma_f32x4(a0, b, acc[0][t]);
        acc[1][t] = wmma_f32x4(a1, b, acc[1][t]);
      }
    }
#pragma unroll
    for (int t = 0; t < 2; ++t) {
      const int col = wave * 32 + t * 16 + lrow;
      const float bia = b2[col];
      const float gg = g2[col] * inv;
      const float ss = s2[col];
#pragma unroll
      for (int rt = 0; rt < 2; ++rt)
#pragma unroll
        for (int r = 0; r < 8; ++r) {
          const int row = row0 + rt * 16 + r + 8 * half;
          if (row < NN) {
            float v = (acc[rt][t][r] + bia) * gg + ss;
            if (relu_out) v = fmaxf(v, 0.0f);
            const size_t idx = (size_t)row * DD + col;
            h[idx] = v + h[idx];  // residual, in-place (element owned by this thread)
          }
        }
    }
  }
}

extern "C" void kernel_launch(void* const* d_in, const int* in_sizes, int n_in,
                              void* d_out, int out_size, void* d_ws, size_t ws_size,
                              hipStream_t stream) {
  (void)in_sizes; (void)n_in; (void)out_size; (void)ws_size;
  const float* x    = (const float*)d_in[0];
  const int*   ei   = (const int*)d_in[1];     // [2, E]
  const int*   ea   = (const int*)d_in[2];     // [E, 3]
  /* d_in[3] = batch (unused in eval-mode reference) */
  const float* eps  = (const float*)d_in[4];   // [L]
  const float* bond = (const float*)d_in[5];   // [L, 3, V, D]
  const float* W1   = (const float*)d_in[6];   // [L, D, 2D]
  const float* b1   = (const float*)d_in[7];
  const float* g1   = (const float*)d_in[8];
  const float* s1   = (const float*)d_in[9];
  const float* W2   = (const float*)d_in[10];  // [L, 2D, D]
  const float* b2   = (const float*)d_in[11];
  const float* g2   = (const float*)d_in[12];
  const float* s2   = (const float*)d_in[13];

  float* h = (float*)d_out;                    // node features, updated in place
  float* z = (float*)d_ws;                     // [N, D] aggregation buffer
  const int* src = ei;
  const int* dst = ei + EE;

  hipMemcpyAsync(h, x, (size_t)NN * DD * sizeof(float),
                 hipMemcpyDeviceToDevice, stream);

  const dim3 b256(256);
  const int initGrid = (NN * DD / 4 + 255) / 256;
  const int edgeGrid = (EE + 7) / 8;
  const int mlpGrid  = (NN + TM - 1) / TM;  // 1563

  for (int l = 0; l < LL; ++l) {
    init_z_kernel<<<initGrid, b256, 0, stream>>>((const float4*)h, eps, l, (float4*)z);
    const float* be = bond + (size_t)l * 3 * VV * DD;
    edge_kernel<<<edgeGrid, b256, 0, stream>>>(h, src, dst, ea,
                                               be, be + VV * DD, be + 2 * VV * DD, z);
    mlp_kernel<<<mlpGrid, b256, 0, stream>>>(z, h,
        W1 + (size_t)l * DD * D2, b1 + l * D2, g1 + l * D2, s1 + l * D2,
        W2 + (size_t)l * D2 * DD, b2 + l * DD, g2 + l * DD, s2 + l * DD,
        (l < LL - 1) ? 1 : 0);
  }
}